// ECELoss_15899968929867
// MI455X (gfx1250) — compile-verified
//
#include <hip/hip_runtime.h>

#define NBINS 15
#define WAVES_PER_BLOCK 8
#define BLOCK_THREADS 256
#define GRID_BLOCKS 2048

// ---------------------------------------------------------------------------
// Kernel 1: zero the 45-float accumulator area in workspace (harness poisons
// d_ws with 0xAA and never re-poisons between graph replays).
// ---------------------------------------------------------------------------
__global__ void ece_zero_kernel(float* __restrict__ bins) {
    int t = threadIdx.x;
    if (t < 3 * NBINS) bins[t] = 0.0f;
}

// ---------------------------------------------------------------------------
// Kernel 2: one wave32 per row. Streaming, memory-bound (~808 MB @ 23.3 TB/s).
//   - 4 coalesced non-temporal b32 loads cover C<=128 columns
//   - 5-step wave32 butterfly for (max, lowest-index argmax) and sum(exp)
//   - per-wave private LDS bins via ds_add_f32 (no contention, fire&forget)
//   - per-block flush: 45 global atomic f32 adds per block
// ---------------------------------------------------------------------------
__global__ __launch_bounds__(BLOCK_THREADS) void ece_main_kernel(
    const float* __restrict__ logits,
    const int*   __restrict__ labels,
    const float* __restrict__ temperature,
    float*       __restrict__ bins,   // [NBINS][3] = {count, sum_conf, sum_acc}
    int N, int C) {

    __shared__ float sbins[WAVES_PER_BLOCK][3 * NBINS];

    const int lane = threadIdx.x & 31;
    const int wib  = threadIdx.x >> 5;   // wave in block

    // zero per-wave private bins
    for (int i = threadIdx.x; i < WAVES_PER_BLOCK * 3 * NBINS; i += BLOCK_THREADS)
        (&sbins[0][0])[i] = 0.0f;
    __syncthreads();

    const float invT    = 1.0f / temperature[0];
    const float NEG_INF = -__builtin_inff();

    const long long rowStride = (long long)gridDim.x * WAVES_PER_BLOCK;
    for (long long row = (long long)blockIdx.x * WAVES_PER_BLOCK + wib;
         row < (long long)N; row += rowStride) {

        const float* rp = logits + row * (long long)C;

        // CDNA5 speculative prefetch (global_prefetch_b8) of this wave's next row;
        // lane*4 element stride -> 32 lanes cover 512B >= one 400B row.
        if (row + rowStride < (long long)N) {
            const float* rpn = logits + (row + rowStride) * (long long)C;
            __builtin_prefetch(rpn + lane * 4, 0, 0);
        }

        // ---- load up to 4 elements per lane (C <= 128), track local argmax ----
        float v[4];
        float bv = NEG_INF;
        int   bi = 0x7fffffff;
#pragma unroll
        for (int k = 0; k < 4; ++k) {
            int idx = (k << 5) + lane;
            v[k] = (idx < C) ? __builtin_nontemporal_load(rp + idx) : NEG_INF;
            if (v[k] > bv) { bv = v[k]; bi = idx; }   // strict > keeps lowest index
        }

        // ---- wave32 butterfly: max with lowest-index tiebreak (matches jnp.argmax)
#pragma unroll
        for (int off = 16; off > 0; off >>= 1) {
            float ov = __shfl_xor(bv, off, 32);
            int   oi = __shfl_xor(bi, off, 32);
            if (ov > bv || (ov == bv && oi < bi)) { bv = ov; bi = oi; }
        }
        const float m = bv;   // all lanes hold row max; bi holds row argmax

        // ---- sum of exp((x - max)/T); inactive slots are -inf -> exp = 0 ----
        float e = 0.0f;
#pragma unroll
        for (int k = 0; k < 4; ++k) e += expf((v[k] - m) * invT);
#pragma unroll
        for (int off = 16; off > 0; off >>= 1) e += __shfl_xor(e, off, 32);

        if (lane == 0) {
            float conf = 1.0f / e;                         // max softmax value
            int b = (int)ceilf(conf * (float)NBINS) - 1;   // (b/n, (b+1)/n] bins
            b = min(max(b, 0), NBINS - 1);
            float acc = (bi == labels[row]) ? 1.0f : 0.0f;
            float* wb = sbins[wib];                        // private -> ds_add_f32
            atomicAdd(&wb[b * 3 + 0], 1.0f);
            atomicAdd(&wb[b * 3 + 1], conf);
            atomicAdd(&wb[b * 3 + 2], acc);
        }
    }

    __syncthreads();
    // flush block partials: 45 global atomic adds per block
    if (threadIdx.x < 3 * NBINS) {
        float s = 0.0f;
#pragma unroll
        for (int w = 0; w < WAVES_PER_BLOCK; ++w) s += sbins[w][threadIdx.x];
        atomicAdd(&bins[threadIdx.x], s);
    }
}

// ---------------------------------------------------------------------------
// Kernel 3: finalize ECE from the 15 bins (one wave).
// Same arithmetic form as the reference (safe_count, gap, prop_in_bin).
// ---------------------------------------------------------------------------
__global__ void ece_final_kernel(const float* __restrict__ bins,
                                 float* __restrict__ out, float invN) {
    int lane = threadIdx.x & 31;
    float g = 0.0f;
    if (lane < NBINS) {
        float cnt  = bins[lane * 3 + 0];
        float sc   = bins[lane * 3 + 1];
        float sa   = bins[lane * 3 + 2];
        float safe = fmaxf(cnt, 1.0f);
        float gap  = (cnt > 0.0f) ? fabsf(sc / safe - sa / safe) : 0.0f;
        g = gap * (cnt * invN);
    }
#pragma unroll
    for (int off = 16; off > 0; off >>= 1) g += __shfl_xor(g, off, 32);
    if (lane == 0) out[0] = g;
}

// ---------------------------------------------------------------------------
extern "C" void kernel_launch(void* const* d_in, const int* in_sizes, int n_in,
                              void* d_out, int out_size, void* d_ws, size_t ws_size,
                              hipStream_t stream) {
    const float* logits      = (const float*)d_in[0];
    const int*   labels      = (const int*)d_in[1];
    const float* temperature = (const float*)d_in[2];
    float*       out         = (float*)d_out;
    float*       bins        = (float*)d_ws;   // 45 floats of scratch

    const int N = in_sizes[1];                 // 2,000,000 rows
    const int C = in_sizes[0] / N;             // 100 classes (kernel supports C<=128)

    ece_zero_kernel<<<1, 64, 0, stream>>>(bins);
    ece_main_kernel<<<GRID_BLOCKS, BLOCK_THREADS, 0, stream>>>(
        logits, labels, temperature, bins, N, C);
    ece_final_kernel<<<1, 32, 0, stream>>>(bins, out, 1.0f / (float)N);
}